// MolecularGraphNeuralNetwork_5119601017308
// MI455X (gfx1250) — compile-verified
//
#include <hip/hip_runtime.h>
#include <hip/hip_bf16.h>

// CDNA5 fp32 WMMA fragment types: 16x16x4 f32 -> A,B = 2 VGPRs (float2), C/D = 8 VGPRs
typedef __attribute__((ext_vector_type(2))) float v2f;
typedef __attribute__((ext_vector_type(8))) float v8f;

#define T_ATOMS  8192
#define N_MOL    256
#define APM      32      // atoms per molecule (T/B)
#define DIM      64
#define L_HID    3
#define L_OUT    2
#define EPS      1e-12f

// One 256-thread block (8 wave32) per molecule. Fully fused GNN:
//   v0 = embed[fp]
//   3x: h = relu(v @ W^T + b); hs = h + A_blk @ h; v = l2norm_rows(hs)
//   mol = sum_atoms(v); 2x MLP; out = mol . W_prop + b_prop
__global__ __launch_bounds__(256) void mgnn_fused_kernel(
    const int*   __restrict__ fingerprints,   // [T]
    const float* __restrict__ adjacency,      // [T,T] block-diagonal
    const float* __restrict__ embed,          // [NVOCAB, DIM]
    const float* __restrict__ W_fp,           // [3, DIM, DIM]
    const float* __restrict__ b_fp,           // [3, DIM]
    const float* __restrict__ W_out,          // [2, DIM, DIM]
    const float* __restrict__ b_out,          // [2, DIM]
    const float* __restrict__ W_prop,         // [1, DIM]
    const float* __restrict__ b_prop,         // [1]
    float*       __restrict__ out)            // [B, 1]
{
    // +1 padding on rows: strided (down-a-column) LDS gathers hit distinct banks
    __shared__ float lds_v[APM][DIM + 1];     // current activations v / hs
    __shared__ float lds_h[APM][DIM + 1];     // h = relu(vW^T+b)
    __shared__ float lds_a[APM][APM + 1];     // molecule's 32x32 adjacency block
    __shared__ float lds_w[DIM][DIM + 1];     // current layer weight (row-major [j][k])
    __shared__ float lds_norm[APM];
    __shared__ float lds_mol[DIM];

    const int molId = blockIdx.x;
    const int tid   = threadIdx.x;
    const int lane  = tid & 31;
    const int wave  = tid >> 5;      // 0..7
    const int l16   = lane & 15;     // N / M position within tile
    const int lh    = lane >> 4;     // half-wave select (K group / +8 rows)
    const int mt    = wave >> 2;     // M tile 0..1  (rows mt*16..)
    const int nt    = wave & 3;      // N tile 0..3  (cols nt*16..)

    // ---- stage v0 = embed[fingerprints] and the diagonal adjacency block ----
    for (int i = tid; i < APM * DIM; i += 256) {
        const int a = i >> 6, d = i & 63;
        const int fp = fingerprints[molId * APM + a];
        lds_v[a][d] = embed[(size_t)fp * DIM + d];
    }
    for (int i = tid; i < APM * APM; i += 256) {
        const int r = i >> 5, c = i & 31;
        lds_a[r][c] = adjacency[(size_t)(molId * APM + r) * T_ATOMS + (molId * APM + c)];
    }
    __syncthreads();

    // ------------------------------ GNN layers ------------------------------
    for (int l = 0; l < L_HID; ++l) {
        // stage this layer's weight into LDS (coalesced)
        for (int i = tid; i < DIM * DIM; i += 256)
            lds_w[i >> 6][i & 63] = W_fp[l * DIM * DIM + i];
        __syncthreads();

        // h = relu(v @ W^T + b): [32x64] = [32x64] x [64x64], K=64 -> 16 wmma steps
        v8f acc = {};
#pragma unroll
        for (int kt = 0; kt < 16; ++kt) {
            const int kb = kt * 4 + lh * 2;
            v2f a, b;
            a.x = lds_v[mt * 16 + l16][kb];
            a.y = lds_v[mt * 16 + l16][kb + 1];
            // B[k][j] = W[j][k] (W^T), j = nt*16 + l16
            b.x = lds_w[nt * 16 + l16][kb];
            b.y = lds_w[nt * 16 + l16][kb + 1];
            acc = __builtin_amdgcn_wmma_f32_16x16x4_f32(
                false, a, false, b, (short)0, acc, false, false);
        }
        {
            const float bias = b_fp[l * DIM + nt * 16 + l16];
#pragma unroll
            for (int r = 0; r < 8; ++r) {
                float h = acc[r] + bias;
                lds_h[mt * 16 + lh * 8 + r][nt * 16 + l16] = h > 0.f ? h : 0.f;
            }
        }
        __syncthreads();

        // hs = h + A_blk @ h: [32x64] += [32x32] x [32x64], K=32 -> 8 wmma steps
        v8f acc2 = {};
#pragma unroll
        for (int kt = 0; kt < 8; ++kt) {
            const int kb = kt * 4 + lh * 2;
            v2f a, b;
            a.x = lds_a[mt * 16 + l16][kb];
            a.y = lds_a[mt * 16 + l16][kb + 1];
            b.x = lds_h[kb][nt * 16 + l16];
            b.y = lds_h[kb + 1][nt * 16 + l16];
            acc2 = __builtin_amdgcn_wmma_f32_16x16x4_f32(
                false, a, false, b, (short)0, acc2, false, false);
        }
#pragma unroll
        for (int r = 0; r < 8; ++r) {
            const int row = mt * 16 + lh * 8 + r;
            const int col = nt * 16 + l16;
            lds_v[row][col] = acc2[r] + lds_h[row][col];
        }
        __syncthreads();

        // row-wise L2 normalize: v = hs / max(||hs||, eps)
        if (tid < APM) {
            float s = 0.f;
#pragma unroll
            for (int d = 0; d < DIM; ++d) { const float x = lds_v[tid][d]; s += x * x; }
            lds_norm[tid] = 1.f / fmaxf(sqrtf(s), EPS);
        }
        __syncthreads();
        for (int i = tid; i < APM * DIM; i += 256)
            lds_v[i >> 6][i & 63] *= lds_norm[i >> 6];
        __syncthreads();
    }

    // ---------------- segment sum -> [DIM] molecular vector ----------------
    if (tid < DIM) {
        float s = 0.f;
#pragma unroll
        for (int a = 0; a < APM; ++a) s += lds_v[a][tid];
        lds_mol[tid] = s;
    }
    __syncthreads();

    // ----------------------------- output MLP ------------------------------
    for (int l = 0; l < L_OUT; ++l) {
        float r = 0.f;
        if (tid < DIM) {
            const float* w = &W_out[(l * DIM + tid) * DIM];
#pragma unroll
            for (int k = 0; k < DIM; ++k) r += lds_mol[k] * w[k];
            r += b_out[l * DIM + tid];
            r = fmaxf(r, 0.f);
        }
        __syncthreads();
        if (tid < DIM) lds_mol[tid] = r;
        __syncthreads();
    }

    if (tid == 0) {
        float s = 0.f;
#pragma unroll
        for (int k = 0; k < DIM; ++k) s += lds_mol[k] * W_prop[k];
        out[molId] = s + b_prop[0];
    }
}

extern "C" void kernel_launch(void* const* d_in, const int* in_sizes, int n_in,
                              void* d_out, int out_size, void* d_ws, size_t ws_size,
                              hipStream_t stream) {
    const int*   fingerprints = (const int*)  d_in[0];
    const float* adjacency    = (const float*)d_in[1];
    // d_in[2] = segment_ids: repeat(arange(B), 32) by construction -> implicit
    const float* embed        = (const float*)d_in[3];
    const float* W_fp         = (const float*)d_in[4];
    const float* b_fp         = (const float*)d_in[5];
    const float* W_out        = (const float*)d_in[6];
    const float* b_out        = (const float*)d_in[7];
    const float* W_prop       = (const float*)d_in[8];
    const float* b_prop       = (const float*)d_in[9];
    float*       out          = (float*)d_out;

    mgnn_fused_kernel<<<N_MOL, 256, 0, stream>>>(
        fingerprints, adjacency, embed, W_fp, b_fp, W_out, b_out, W_prop, b_prop, out);
}